// S2VT_Net_53523882443061
// MI455X (gfx1250) — compile-verified
//
#include <hip/hip_runtime.h>
#include <hip/hip_bf16.h>
#include <math.h>

// ---------------------------------------------------------------------------
// Problem sizes (fixed by the reference)
// ---------------------------------------------------------------------------
#define BB 128
#define TT 80
#define DD 4096
#define HH 1024
#define VV 6000
#define SS 10
#define GG (3 * HH)

typedef float v2f __attribute__((ext_vector_type(2)));
typedef float v8f __attribute__((ext_vector_type(8)));

#define KC 32          // K-chunk staged per barrier pair
#define LDS_STRIDE 34  // padded row stride (floats): even (b64-aligned),
                       // 34 = 2*17, 17 coprime with 32 -> conflict-free

// ---------------------------------------------------------------------------
// WMMA f32 GEMM:  C[M,N] = act( A[M,K](lda) * W[N,K](ldw)^T + bias[N] )
// Block: 256 threads (8 wave32), 128x64 macro tile.
// Wave w: rg = w>>1 (32 rows), cg = w&1 (32 cols) -> 32x32 per wave,
// 4 accumulators, 32 v_wmma per K-chunk per wave.
// Software pipeline: next K-chunk is global-loaded into registers while the
// current chunk's WMMAs execute, then stored to LDS.
// Requires M % 128 == 0, K % 32 == 0 (true for all call sites); N guarded
// only in the NGUARD=true instantiation (logits GEMM, N=6000).
// ---------------------------------------------------------------------------
template <bool NGUARD>
__global__ __launch_bounds__(256) void wmma_gemm_bias(
    const float* __restrict__ A, long lda,
    const float* __restrict__ W, long ldw,
    const float* __restrict__ bias,
    float* __restrict__ C, long ldc,
    int M, int N, int K, int act) {
  __shared__ float As[128 * LDS_STRIDE];
  __shared__ float Ws[64 * LDS_STRIDE];

  const int tid  = threadIdx.x;
  const int lane = tid & 31;
  const int wave = tid >> 5;
  const int rg   = wave >> 1;  // 0..3 (32-row group)
  const int cg   = wave & 1;   // 0..1 (32-col group)

  const int nblk = blockIdx.x * 64;
  const int mblk = blockIdx.y * 128;

  // staging assignment
  const int arow = tid >> 1;         // 0..127
  const int acol = (tid & 1) * 16;   // 0 or 16  (4x float4)
  const int wrow = tid >> 2;         // 0..63
  const int wcol = (tid & 3) * 8;    // 0..24    (2x float4)

  const int ml = lane & 15;        // M / N index within fragment
  const int kh = (lane >> 4) * 2;  // K offset within fragment

  v8f acc00 = {0.f, 0.f, 0.f, 0.f, 0.f, 0.f, 0.f, 0.f};
  v8f acc01 = acc00, acc10 = acc00, acc11 = acc00;

  const float* aPtr = A + (long)(mblk + arow) * lda + acol;
  const int nrow = nblk + wrow;
  const float* wPtr = W + (long)nrow * ldw + wcol;
  const bool wOK = !NGUARD || (nrow < N);

  float4 aR[4], wR[2];
  // prologue: load chunk 0 into registers
#pragma unroll
  for (int i = 0; i < 4; ++i) aR[i] = *(const float4*)(aPtr + i * 4);
#pragma unroll
  for (int i = 0; i < 2; ++i) {
    wR[i] = make_float4(0.f, 0.f, 0.f, 0.f);
    if (wOK) wR[i] = *(const float4*)(wPtr + i * 4);
  }

  for (int k0 = 0; k0 < K; k0 += KC) {
    if (k0) __syncthreads();  // prior chunk's fragment reads complete
#pragma unroll
    for (int i = 0; i < 4; ++i) {
      As[arow * LDS_STRIDE + acol + i * 4 + 0] = aR[i].x;
      As[arow * LDS_STRIDE + acol + i * 4 + 1] = aR[i].y;
      As[arow * LDS_STRIDE + acol + i * 4 + 2] = aR[i].z;
      As[arow * LDS_STRIDE + acol + i * 4 + 3] = aR[i].w;
    }
#pragma unroll
    for (int i = 0; i < 2; ++i) {
      Ws[wrow * LDS_STRIDE + wcol + i * 4 + 0] = wR[i].x;
      Ws[wrow * LDS_STRIDE + wcol + i * 4 + 1] = wR[i].y;
      Ws[wrow * LDS_STRIDE + wcol + i * 4 + 2] = wR[i].z;
      Ws[wrow * LDS_STRIDE + wcol + i * 4 + 3] = wR[i].w;
    }
    __syncthreads();

    // register-prefetch the NEXT chunk; latency hides under the WMMAs below
    if (k0 + KC < K) {
#pragma unroll
      for (int i = 0; i < 4; ++i)
        aR[i] = *(const float4*)(aPtr + k0 + KC + i * 4);
#pragma unroll
      for (int i = 0; i < 2; ++i)
        if (wOK) wR[i] = *(const float4*)(wPtr + k0 + KC + i * 4);
      if (k0 + 2 * KC < K)  // L2 prefetch two chunks ahead
        __builtin_prefetch(aPtr + k0 + 2 * KC, 0, 0);
    }

#pragma unroll
    for (int kk = 0; kk < KC; kk += 4) {
      v2f a0 = *(const v2f*)&As[(rg * 32 + ml) * LDS_STRIDE + kk + kh];
      v2f a1 = *(const v2f*)&As[(rg * 32 + 16 + ml) * LDS_STRIDE + kk + kh];
      v2f b0 = *(const v2f*)&Ws[(cg * 32 + ml) * LDS_STRIDE + kk + kh];
      v2f b1 = *(const v2f*)&Ws[(cg * 32 + 16 + ml) * LDS_STRIDE + kk + kh];
      acc00 = __builtin_amdgcn_wmma_f32_16x16x4_f32(
          false, a0, false, b0, (short)0, acc00, false, false);
      acc01 = __builtin_amdgcn_wmma_f32_16x16x4_f32(
          false, a0, false, b1, (short)0, acc01, false, false);
      acc10 = __builtin_amdgcn_wmma_f32_16x16x4_f32(
          false, a1, false, b0, (short)0, acc10, false, false);
      acc11 = __builtin_amdgcn_wmma_f32_16x16x4_f32(
          false, a1, false, b1, (short)0, acc11, false, false);
    }
  }

  const int n0 = nblk + cg * 32 + ml;
  const int n1 = n0 + 16;
  float bb0 = 0.f, bb1 = 0.f;
  if (bias != nullptr) {
    if (!NGUARD || n0 < N) bb0 = bias[n0];
    if (!NGUARD || n1 < N) bb1 = bias[n1];
  }
  const int m0 = mblk + rg * 32 + (lane >> 4) * 8;

#pragma unroll
  for (int v = 0; v < 8; ++v) {
    float c00 = acc00[v] + bb0;
    float c01 = acc01[v] + bb1;
    float c10 = acc10[v] + bb0;
    float c11 = acc11[v] + bb1;
    if (act) {
      c00 = fmaxf(c00, 0.f); c01 = fmaxf(c01, 0.f);
      c10 = fmaxf(c10, 0.f); c11 = fmaxf(c11, 0.f);
    }
    if (!NGUARD || n0 < N) {
      C[(long)(m0 + v) * ldc + n0] = c00;
      C[(long)(m0 + 16 + v) * ldc + n0] = c10;
    }
    if (!NGUARD || n1 < N) {
      C[(long)(m0 + v) * ldc + n1] = c01;
      C[(long)(m0 + 16 + v) * ldc + n1] = c11;
    }
  }
}

// ---------------------------------------------------------------------------
// GRU gate elementwise:  h = (1-z)*n + z*h
// gi_stride == 0 -> gi is a bias vector broadcast over batch (decoder eh path)
// ---------------------------------------------------------------------------
__global__ void gru_elem(const float* __restrict__ gi, long gi_stride,
                         const float* __restrict__ gh,
                         float* __restrict__ h, int H) {
  const int b = blockIdx.y;
  const int j = blockIdx.x * blockDim.x + threadIdx.x;
  if (j >= H) return;
  float gir, giz, gin;
  if (gi_stride == 0) {
    gir = gi[j]; giz = gi[H + j]; gin = gi[2 * H + j];
  } else {
    const float* gib = gi + (long)b * gi_stride;
    gir = gib[j]; giz = gib[H + j]; gin = gib[2 * H + j];
  }
  const float* ghb = gh + (long)b * (3 * H);
  const float r = 1.f / (1.f + expf(-(gir + ghb[j])));
  const float z = 1.f / (1.f + expf(-(giz + ghb[H + j])));
  const float n = tanhf(gin + r * ghb[2 * H + j]);
  const long hi = (long)b * H + j;
  h[hi] = (1.f - z) * n + z * h[hi];
}

// xcat[b] = [ eh[b] , embed[words[b]] ]   (B x 2H)
__global__ void build_xcat(const float* __restrict__ eh,
                           const float* __restrict__ embed,
                           const int* __restrict__ words,
                           float* __restrict__ xcat, int H) {
  const int b = blockIdx.y;
  const int j = blockIdx.x * blockDim.x + threadIdx.x;
  if (j >= H) return;
  xcat[(long)b * 2 * H + j] = eh[(long)b * H + j];
  xcat[(long)b * 2 * H + H + j] = embed[(long)words[b] * H + j];
}

__global__ void init_state(float* eh, float* dh, int* words, int BH, int B) {
  const int i = blockIdx.x * blockDim.x + threadIdx.x;
  if (i < BH) { eh[i] = 0.f; dh[i] = 0.f; }
  if (i < B) words[i] = 1;
}

// Per-row log-softmax target loss + argmax (first-occurrence ties).
__global__ __launch_bounds__(256) void softmax_loss(
    const float* __restrict__ logits, const int* __restrict__ target,
    int t, int S, int V, float* __restrict__ rowloss,
    int* __restrict__ words, float invB) {
  __shared__ float sval[256];
  __shared__ int   sidx[256];
  __shared__ float ssum[256];
  const int b = blockIdx.x;
  const int tid = threadIdx.x;
  const float* row = logits + (long)b * V;

  float vmax = -3.402823466e38f;
  int imax = 0;
  for (int i = tid; i < V; i += 256) {
    const float v = row[i];
    if (v > vmax) { vmax = v; imax = i; }
  }
  sval[tid] = vmax; sidx[tid] = imax;
  __syncthreads();
  for (int s = 128; s > 0; s >>= 1) {
    if (tid < s) {
      const float v2 = sval[tid + s];
      const int i2 = sidx[tid + s];
      if (v2 > sval[tid] || (v2 == sval[tid] && i2 < sidx[tid])) {
        sval[tid] = v2; sidx[tid] = i2;
      }
    }
    __syncthreads();
  }
  vmax = sval[0];

  float acc = 0.f;
  for (int i = tid; i < V; i += 256) acc += expf(row[i] - vmax);
  ssum[tid] = acc;
  __syncthreads();
  for (int s = 128; s > 0; s >>= 1) {
    if (tid < s) ssum[tid] += ssum[tid + s];
    __syncthreads();
  }
  if (tid == 0) {
    const int tgt = target[b * S + t];
    const float lp = row[tgt] - vmax - logf(ssum[0]);
    rowloss[b] = -lp * invB;
    words[b] = sidx[0];
  }
}

// Deterministic batch-loss reduction + emit sentence tokens for this step.
__global__ void finalize_step(const float* __restrict__ rowloss,
                              const int* __restrict__ words,
                              float* __restrict__ losses, int t,
                              float* __restrict__ out) {
  __shared__ float s[128];
  const int tid = threadIdx.x;
  s[tid] = rowloss[tid];
  __syncthreads();
  for (int k = 64; k > 0; k >>= 1) {
    if (tid < k) s[tid] += s[tid + k];
    __syncthreads();
  }
  if (tid == 0) {
    losses[t] = s[0];
    out[1 + t]  = (float)words[0];
    out[11 + t] = (float)words[1];
  }
}

__global__ void final_sum(const float* __restrict__ losses,
                          float* __restrict__ out, int S) {
  if (threadIdx.x == 0 && blockIdx.x == 0) {
    float a = 0.f;
    for (int i = 0; i < S; ++i) a += losses[i];
    out[0] = a;
  }
}

// ---------------------------------------------------------------------------
// Host side
// ---------------------------------------------------------------------------
static inline void launch_gemm(const float* A, long lda, const float* W,
                               long ldw, const float* bias, float* C, long ldc,
                               int M, int N, int K, int act, hipStream_t s) {
  dim3 grid((N + 63) / 64, (M + 127) / 128);
  if (N & 63) {
    wmma_gemm_bias<true><<<grid, 256, 0, s>>>(A, lda, W, ldw, bias, C, ldc,
                                              M, N, K, act);
  } else {
    wmma_gemm_bias<false><<<grid, 256, 0, s>>>(A, lda, W, ldw, bias, C, ldc,
                                               M, N, K, act);
  }
}

extern "C" void kernel_launch(void* const* d_in, const int* in_sizes, int n_in,
                              void* d_out, int out_size, void* d_ws,
                              size_t ws_size, hipStream_t stream) {
  (void)in_sizes; (void)n_in; (void)out_size;
  const float* data    = (const float*)d_in[0];
  // d_in[1] = mask (unused by reference math), d_in[3] = sent_len scalar
  const int*   target  = (const int*)d_in[2];
  const float* enc_Wih = (const float*)d_in[4];
  const float* enc_Whh = (const float*)d_in[5];
  const float* enc_bih = (const float*)d_in[6];
  const float* enc_bhh = (const float*)d_in[7];
  const float* dec_Wih = (const float*)d_in[8];
  const float* dec_Whh = (const float*)d_in[9];
  const float* dec_bih = (const float*)d_in[10];
  const float* dec_bhh = (const float*)d_in[11];
  const float* lin1_W  = (const float*)d_in[12];
  const float* lin1_b  = (const float*)d_in[13];
  const float* lin2_W  = (const float*)d_in[14];
  const float* lin2_b  = (const float*)d_in[15];
  const float* embed   = (const float*)d_in[16];
  float* out = (float*)d_out;

  // workspace layout (floats)
  float* ws = (float*)d_ws;
  float* eh      = ws;                       // B*H
  float* dh      = eh + BB * HH;             // B*H
  float* gi      = dh + BB * HH;             // B*3H
  float* gh      = gi + BB * GG;             // B*3H
  float* u       = gh + BB * GG;             // B*H
  float* xcat    = u + BB * HH;              // B*2H
  float* logits  = xcat + BB * 2 * HH;       // B*V
  float* rowloss = logits + (long)BB * VV;   // B
  float* losses  = rowloss + BB;             // 16
  int*   words   = (int*)(losses + 16);      // B ints
  float* GIall   = (float*)(words + BB);     // B*T*3H (optional fast path)

  const size_t base_bytes = (size_t)((char*)GIall - (char*)d_ws);
  const size_t gi_bytes = (size_t)BB * TT * GG * sizeof(float);
  const bool bigGI = (ws_size >= base_bytes + gi_bytes);

  init_state<<<dim3((BB * HH + 255) / 256), 256, 0, stream>>>(
      eh, dh, words, BB * HH, BB);

  // Batched input projection for ALL timesteps: GI = data @ enc_Wih^T + bih
  if (bigGI) {
    launch_gemm(data, DD, enc_Wih, DD, enc_bih, GIall, GG,
                BB * TT, GG, DD, 0, stream);
  }

  const dim3 egrid((HH + 255) / 256, BB);

  // -------------------- encoder loop --------------------
  for (int t = 0; t < TT; ++t) {
    const float* gi_ptr;
    long gi_stride;
    if (bigGI) {
      gi_ptr = GIall + (long)t * GG;   // row b at (b*T + t)*3H
      gi_stride = (long)TT * GG;
    } else {
      launch_gemm(data + (long)t * DD, (long)TT * DD, enc_Wih, DD, enc_bih,
                  gi, GG, BB, GG, DD, 0, stream);
      gi_ptr = gi;
      gi_stride = GG;
    }
    launch_gemm(eh, HH, enc_Whh, HH, enc_bhh, gh, GG, BB, GG, HH, 0, stream);
    gru_elem<<<egrid, 256, 0, stream>>>(gi_ptr, gi_stride, gh, eh, HH);
    // u = relu(eh @ W1a^T + lin1_b), W1a = lin1_W[:, :H] (row stride 2H)
    launch_gemm(eh, HH, lin1_W, 2 * HH, lin1_b, u, HH, BB, HH, HH, 1, stream);
    launch_gemm(u, HH, dec_Wih, HH, dec_bih, gi, GG, BB, GG, HH, 0, stream);
    launch_gemm(dh, HH, dec_Whh, HH, dec_bhh, gh, GG, BB, GG, HH, 0, stream);
    gru_elem<<<egrid, 256, 0, stream>>>(gi, GG, gh, dh, HH);
  }

  // -------------------- decoder loop --------------------
  const float invB = 1.0f / (float)BB;
  for (int t = 0; t < SS; ++t) {
    // eh update: zero-input GRU -> gi is just enc_bih broadcast
    launch_gemm(eh, HH, enc_Whh, HH, enc_bhh, gh, GG, BB, GG, HH, 0, stream);
    gru_elem<<<egrid, 256, 0, stream>>>(enc_bih, 0L, gh, eh, HH);
    // u = relu([eh, embed[word]] @ lin1_W^T + lin1_b)
    build_xcat<<<egrid, 256, 0, stream>>>(eh, embed, words, xcat, HH);
    launch_gemm(xcat, 2 * HH, lin1_W, 2 * HH, lin1_b, u, HH,
                BB, HH, 2 * HH, 1, stream);
    // dec GRU
    launch_gemm(u, HH, dec_Wih, HH, dec_bih, gi, GG, BB, GG, HH, 0, stream);
    launch_gemm(dh, HH, dec_Whh, HH, dec_bhh, gh, GG, BB, GG, HH, 0, stream);
    gru_elem<<<egrid, 256, 0, stream>>>(gi, GG, gh, dh, HH);
    // logits + log-softmax loss + argmax
    launch_gemm(dh, HH, lin2_W, HH, lin2_b, logits, VV, BB, VV, HH, 0, stream);
    softmax_loss<<<dim3(BB), 256, 0, stream>>>(logits, target, t, SS, VV,
                                               rowloss, words, invB);
    finalize_step<<<dim3(1), 128, 0, stream>>>(rowloss, words, losses, t, out);
  }
  final_sum<<<dim3(1), 32, 0, stream>>>(losses, out, SS);
}